// FastOrthogonal_10960756540137
// MI455X (gfx1250) — compile-verified
//
#include <hip/hip_runtime.h>

// CholeskyQR formulation of the reference MGS orthogonalization.
//   X : [M=131072, R=64, 2] float32 (complex interleaved)
//   G = X^H X           (64x64 complex, WMMA f32 Gram reduction)
//   L = chol(G), W = L^{-H}   (one workgroup, LDS)
//   Q = X W             (WMMA f32 tall-skinny GEMM)

#define M_ROWS 131072
#define R_COLS 64

typedef __attribute__((ext_vector_type(2))) float v2f;
typedef __attribute__((ext_vector_type(4))) float v4f;
typedef __attribute__((ext_vector_type(8))) float v8f;

// Workspace layout (bytes):
//   [0      , 16384)  Gr  : float[64][64]  real part of Gram
//   [16384  , 32768)  Gi  : float[64][64]  imag part of Gram
//   [32768  , 65536)  W   : float[64][64][2]  complex L^{-H}
#define WS_GR 0
#define WS_GI 4096   // float offset
#define WS_W  8192   // float offset

__device__ inline v8f wmma_f32x4(v2f a, v2f b, v8f c) {
  // V_WMMA_F32_16X16X4_F32 : D(16x16 f32) = A(16x4 f32) x B(4x16 f32) + C
  return __builtin_amdgcn_wmma_f32_16x16x4_f32(
      /*neg_a=*/false, a, /*neg_b=*/false, b,
      /*c_mod=*/(short)0, c, /*reuse_a=*/false, /*reuse_b=*/false);
}

// ---------------------------------------------------------------- zero Gram
__global__ void fo_zero_gram(float* ws) {
  int i = blockIdx.x * blockDim.x + threadIdx.x;
  if (i < 2 * 64 * 64) ws[i] = 0.0f;
}

// ---------------------------------------------------------------- Gram: G = X^H X
// grid 512 blocks x 512 threads (16 waves). Each block reduces 256 rows.
// wave w -> output tile (ti = w>>2, tj = w&3); holds Gr,Gi 16x16 f32 tiles.
__global__ __launch_bounds__(512) void fo_gram(const float* __restrict__ x,
                                               float* __restrict__ ws) {
  const int lane = threadIdx.x & 31;
  const int wave = threadIdx.x >> 5;
  const int ti = wave >> 2;
  const int tj = wave & 3;
  const int p = lane & 15;              // tile row/col within 16
  const int kb = (lane < 16) ? 0 : 2;   // K rows held by this lane half

  const int colA = 16 * ti + p;
  const int colB = 16 * tj + p;
  const int row0 = blockIdx.x * 256;

  v8f gr = {};
  v8f gi = {};

  for (int ks = 0; ks < 256; ks += 4) {
    const size_t m = (size_t)(row0 + ks + kb);
    // A fragment: A[p][k] = X[m+k][16ti+p]  (transposed slice), (re,im) per b64
    v2f ea0 = *(const v2f*)(x + (m * R_COLS + colA) * 2);
    v2f ea1 = *(const v2f*)(x + ((m + 1) * R_COLS + colA) * 2);
    v2f eb0 = *(const v2f*)(x + (m * R_COLS + colB) * 2);
    v2f eb1 = *(const v2f*)(x + ((m + 1) * R_COLS + colB) * 2);

    v2f ar = {ea0.x, ea1.x};
    v2f ai = {ea0.y, ea1.y};
    v2f ain = {-ea0.y, -ea1.y};
    v2f br = {eb0.x, eb1.x};
    v2f bi = {eb0.y, eb1.y};

    // Gr += Ar^T Br + Ai^T Bi ; Gi += Ar^T Bi - Ai^T Br
    gr = wmma_f32x4(ar, br, gr);
    gr = wmma_f32x4(ai, bi, gr);
    gi = wmma_f32x4(ar, bi, gi);
    gi = wmma_f32x4(ain, br, gi);
  }

  // C/D layout: VGPR v -> row (v + 8*(lane>=16)), col = lane&15
  const int rbase = 16 * ti + ((lane < 16) ? 0 : 8);
  const int c = 16 * tj + p;
  float* grp = ws + WS_GR;
  float* gip = ws + WS_GI;
#pragma unroll
  for (int v = 0; v < 8; ++v) {
    atomicAdd(&grp[(rbase + v) * 64 + c], gr[v]);
    atomicAdd(&gip[(rbase + v) * 64 + c], gi[v]);
  }
}

// ---------------------------------------------------------------- Cholesky + W = L^{-H}
// One block, 64 threads. G, Y in LDS.
__global__ __launch_bounds__(64) void fo_chol_inv(float* __restrict__ ws) {
  __shared__ float2 Gs[64 * 64];  // 32 KB
  __shared__ float2 Ys[64 * 64];  // 32 KB
  const int t = threadIdx.x;
  const float* grp = ws + WS_GR;
  const float* gip = ws + WS_GI;

  for (int i = t; i < 64 * 64; i += 64) {
    Gs[i].x = grp[i];
    Gs[i].y = gip[i];
  }
  __syncthreads();

  // In-place complex Cholesky (lower triangle), G = L L^H
  for (int k = 0; k < 64; ++k) {
    if (t == k) {
      float d = sqrtf(Gs[k * 64 + k].x + 1e-8f);
      Gs[k * 64 + k].x = d;
      Gs[k * 64 + k].y = 0.0f;
    }
    __syncthreads();
    const float d = Gs[k * 64 + k].x;
    if (t > k) {
      Gs[t * 64 + k].x /= d;
      Gs[t * 64 + k].y /= d;
    }
    __syncthreads();
    if (t > k) {
      const float2 ltk = Gs[t * 64 + k];
      for (int j = k + 1; j <= t; ++j) {
        const float2 ljk = Gs[j * 64 + k];  // conj applied below
        Gs[t * 64 + j].x -= ltk.x * ljk.x + ltk.y * ljk.y;
        Gs[t * 64 + j].y -= ltk.y * ljk.x - ltk.x * ljk.y;
      }
    }
    __syncthreads();
  }

  // Minv = L^{-1}: thread c solves L y = e_c (forward substitution).
  const int cidx = t;
  for (int r = 0; r < 64; ++r) {
    float2 s;
    s.x = (r == cidx) ? 1.0f : 0.0f;
    s.y = 0.0f;
    if (r < cidx) {
      Ys[r * 64 + cidx] = make_float2(0.0f, 0.0f);
      continue;
    }
    for (int kk = cidx; kk < r; ++kk) {
      const float2 l = Gs[r * 64 + kk];
      const float2 y = Ys[kk * 64 + cidx];
      s.x -= l.x * y.x - l.y * y.y;
      s.y -= l.x * y.y + l.y * y.x;
    }
    const float d = Gs[r * 64 + r].x;
    s.x /= d;
    s.y /= d;
    Ys[r * 64 + cidx] = s;
  }
  __syncthreads();

  // W = Minv^H : W[c][j] = conj(Minv[j][c]) ; thread c writes row c of W.
  float* wp = ws + WS_W;
  for (int j = 0; j < 64; ++j) {
    const float2 y = Ys[j * 64 + cidx];
    wp[(cidx * 64 + j) * 2 + 0] = y.x;
    wp[(cidx * 64 + j) * 2 + 1] = -y.y;
  }
}

// ---------------------------------------------------------------- Apply: Q = X W
// grid 2048 blocks x 512 threads (16 waves); block covers 64 rows.
// wave w -> (row strip s = w>>2, col tile tj = w&3). W cached in LDS.
__global__ __launch_bounds__(512) void fo_apply(const float* __restrict__ x,
                                                const float* __restrict__ ws,
                                                float* __restrict__ out) {
  __shared__ float2 Wl[64 * 64];  // 32 KB: W complex
  const float* wp = ws + WS_W;
  for (int i = threadIdx.x; i < 64 * 64; i += 512) {
    Wl[i].x = wp[2 * i + 0];
    Wl[i].y = wp[2 * i + 1];
  }
  __syncthreads();

  const int lane = threadIdx.x & 31;
  const int wave = threadIdx.x >> 5;
  const int s = wave >> 2;
  const int tj = wave & 3;
  const int p = lane & 15;
  const int kb = (lane < 16) ? 0 : 2;

  const size_t row0 = (size_t)blockIdx.x * 64 + 16 * s;

  v8f qr = {};
  v8f qi = {};

  for (int ks = 0; ks < 64; ks += 4) {
    // A fragment: A[p][k] = X[row0+p][ks+k]; cols (kb,kb+1) re/im -> one b128
    v4f e = *(const v4f*)(x + ((row0 + p) * R_COLS + ks + kb) * 2);
    v2f ar = {e.x, e.z};
    v2f ai = {e.y, e.w};
    v2f ain = {-e.y, -e.w};

    // B fragment: B[k][q] = W[ks+k][16tj+q]
    const float2 w0 = Wl[(ks + kb) * 64 + 16 * tj + p];
    const float2 w1 = Wl[(ks + kb + 1) * 64 + 16 * tj + p];
    v2f wr = {w0.x, w1.x};
    v2f wi = {w0.y, w1.y};

    // Qr += Ar Wr - Ai Wi ; Qi += Ar Wi + Ai Wr
    qr = wmma_f32x4(ar, wr, qr);
    qr = wmma_f32x4(ain, wi, qr);
    qi = wmma_f32x4(ar, wi, qi);
    qi = wmma_f32x4(ai, wr, qi);
  }

  const int rb = (lane < 16) ? 0 : 8;
  const int c = 16 * tj + p;
#pragma unroll
  for (int v = 0; v < 8; ++v) {
    const size_t row = row0 + rb + v;
    v2f o = {qr[v], qi[v]};
    *(v2f*)(out + (row * R_COLS + c) * 2) = o;
  }
}

// ---------------------------------------------------------------- launch
extern "C" void kernel_launch(void* const* d_in, const int* in_sizes, int n_in,
                              void* d_out, int out_size, void* d_ws, size_t ws_size,
                              hipStream_t stream) {
  (void)in_sizes; (void)n_in; (void)out_size; (void)ws_size;
  const float* x = (const float*)d_in[0];
  float* out = (float*)d_out;
  float* ws = (float*)d_ws;

  fo_zero_gram<<<(2 * 64 * 64 + 255) / 256, 256, 0, stream>>>(ws);
  fo_gram<<<M_ROWS / 256, 512, 0, stream>>>(x, ws);
  fo_chol_inv<<<1, 64, 0, stream>>>(ws);
  fo_apply<<<M_ROWS / 64, 512, 0, stream>>>(x, ws, out);
}